// radar_Dymap_90950227460802
// MI455X (gfx1250) — compile-verified
//
#include <hip/hip_runtime.h>
#include <stdint.h>

#define BATCH   4
#define NPTS    1000000
#define NYD     512
#define NXD     512
#define CANVAS  (NYD * NXD)          // 262144 per batch
#define TOTAL_PTS (BATCH * NPTS)     // 4,000,000
#define PPT     4                    // points per thread
#define TPB     256                  // 8 wave32 per block

typedef unsigned uv4 __attribute__((ext_vector_type(4)));
typedef float    fv4 __attribute__((ext_vector_type(4)));

// ---------------------------------------------------------------------------
// Kernel 0: zero the canvas (d_out reused as u32 canvas). 1,048,576 u32 total,
// 4 per thread via b128 stores -> exactly 1024 blocks x 256 threads.
// ---------------------------------------------------------------------------
__global__ __launch_bounds__(TPB) void zero_canvas_kernel(uv4* __restrict__ out) {
    const unsigned t = blockIdx.x * TPB + threadIdx.x;
    out[t] = (uv4){0u, 0u, 0u, 0u};
}

// ---------------------------------------------------------------------------
// Kernel 1: scatter-max. Each thread handles 4 consecutive points (NPTS is a
// multiple of 4, so all 4 land in the same batch).
//  - |p4| stream staged GMEM->LDS with gfx1250 async copies (ASYNCcnt path)
//  - coords read as 3x b128 non-temporal loads (12 contiguous ints/thread,
//    16B aligned) so the streamed 128 MB doesn't evict the L2-resident canvas
//  - per-voxel max as u32 atomic on the IEEE bit pattern (valid: |p4| >= 0,
//    canvas zero-initialized, non-negative floats order as their u32 bits)
// ---------------------------------------------------------------------------
__global__ __launch_bounds__(TPB) void scatter_max_kernel(
        const float* __restrict__ pts,      // [B, N, 5]
        const int*   __restrict__ coords,   // [B, N, 3] (z, y, x)
        unsigned*    __restrict__ canvas)   // [B, NY*NX] as u32 bits
{
    __shared__ unsigned stage[TPB * PPT];

    const long long t    = (long long)blockIdx.x * TPB + threadIdx.x;
    const long long base = t * PPT;
    if (base >= TOTAL_PTS) return;

    // LDS byte offset of this thread's staging slot: generic shared-memory
    // addresses carry the LDS offset in their low 32 bits.
    const unsigned lds0 = (unsigned)(uintptr_t)&stage[threadIdx.x * PPT];

    // --- async-stage p[n][4] for 4 points: GMEM -> LDS (ASYNCcnt) ---
    // GVS addressing: SADDR(64) + VADDR_I32; byte offset = n*20 + 16 < 2^31.
#pragma unroll
    for (int i = 0; i < PPT; ++i) {
        const unsigned goff = (unsigned)((base + i) * 20 + 16);
        asm volatile("global_load_async_to_lds_b32 %0, %1, %2"
                     :
                     : "v"(lds0 + 4u * (unsigned)i), "v"(goff), "s"(pts)
                     : "memory");
    }

    // --- coords: 12 contiguous ints = 3x b128 NT loads, swizzle out y/x ---
    const uv4* __restrict__ c4 = (const uv4*)coords;   // 16B-aligned chunks
    const long long cw = t * 3;                        // (base*3)/4
    const uv4 c0 = __builtin_nontemporal_load(&c4[cw + 0]); // z0 y0 x0 z1
    const uv4 c1 = __builtin_nontemporal_load(&c4[cw + 1]); // y1 x1 z2 y2
    const uv4 c2 = __builtin_nontemporal_load(&c4[cw + 2]); // x2 z3 y3 x3

    unsigned ys[PPT] = {c0.y, c1.x, c1.w, c2.z};
    unsigned xs[PPT] = {c0.z, c1.y, c2.x, c2.w};

    const unsigned boff = (unsigned)(base / NPTS) * (unsigned)CANVAS;

    // Fence our async copies, then consume the staged values from LDS.
    asm volatile("s_wait_asynccnt 0" ::: "memory");

#pragma unroll
    for (int i = 0; i < PPT; ++i) {
        const unsigned bits = stage[threadIdx.x * PPT + i] & 0x7fffffffu; // |p4|
        const unsigned idx  = boff + ys[i] * NXD + xs[i];
        atomicMax(&canvas[idx], bits);   // global_atomic_max_u32 (no return)
    }
}

// ---------------------------------------------------------------------------
// Kernel 2: in-place threshold. Each thread owns one uv4 of the canvas
// (race-free), converts bit patterns back to float and writes
// (v > 0.1f) ? 1 : 0 with a non-temporal b128 store (write-once data).
// ---------------------------------------------------------------------------
__global__ __launch_bounds__(TPB) void threshold_kernel(float* __restrict__ out) {
    const unsigned t = blockIdx.x * TPB + threadIdx.x;
    const uv4 v = ((const uv4*)out)[t];
    fv4 r;
    r.x = (__uint_as_float(v.x) > 0.1f) ? 1.0f : 0.0f;
    r.y = (__uint_as_float(v.y) > 0.1f) ? 1.0f : 0.0f;
    r.z = (__uint_as_float(v.z) > 0.1f) ? 1.0f : 0.0f;
    r.w = (__uint_as_float(v.w) > 0.1f) ? 1.0f : 0.0f;
    __builtin_nontemporal_store(r, &((fv4*)out)[t]);
}

extern "C" void kernel_launch(void* const* d_in, const int* in_sizes, int n_in,
                              void* d_out, int out_size, void* d_ws, size_t ws_size,
                              hipStream_t stream) {
    (void)in_sizes; (void)n_in; (void)out_size; (void)d_ws; (void)ws_size;

    const float* pts    = (const float*)d_in[0];   // [4, 1e6, 5] f32
    const int*   coords = (const int*)d_in[1];     // [4, 1e6, 3] i32
    unsigned*    canvas = (unsigned*)d_out;        // reused as u32 canvas

    // 1,048,576 canvas elems / 4 per thread / 256 threads = 1024 blocks exact.
    const int canvas_blocks = (BATCH * CANVAS) / 4 / TPB;
    zero_canvas_kernel<<<canvas_blocks, TPB, 0, stream>>>((uv4*)d_out);

    const int nthreads = TOTAL_PTS / PPT;          // 1,000,000
    const int sblocks  = (nthreads + TPB - 1) / TPB;
    scatter_max_kernel<<<sblocks, TPB, 0, stream>>>(pts, coords, canvas);

    threshold_kernel<<<canvas_blocks, TPB, 0, stream>>>((float*)d_out);
}